// GCN_59674275610923
// MI455X (gfx1250) — compile-verified
//
#include <hip/hip_runtime.h>

typedef __attribute__((ext_vector_type(2))) float v2f;
typedef __attribute__((ext_vector_type(8))) float v8f;

struct alignas(16) LL2 { long long x, y; };

// ---------------------------------------------------------------------------
// Kernel A: per-node MLP  h[i] = b2 + sum_k W2[k] * relu(x[i]*W1[k] + b1[k])
// via V_WMMA_F32_16X16X4_F32 (K=16 as 4 accumulating WMMAs per 16-node tile).
// Also zeroes the interleaved (agg, cnt) accumulators that live in d_out.
//
// Layout (ISA 7.12.2, f32 shapes):
//   A 16x4:  VGPR0 = {K0 lanes0-15, K2 lanes16-31}, VGPR1 = {K1, K3}
//   B 4x16:  VGPR0 = {row K0 lanes0-15, row K2 lanes16-31}, VGPR1 = {K1, K3}
//   -> per lane: kb = 4*c + 2*(lane>>4) selects the two K slots it owns.
// B[k][n] = W2[k] broadcast over all n  =>  D[m][n] == h_pre[m] for every n,
// so only the documented D row<->VGPR mapping matters for the store:
//   D vgpr j, lanes0-15: M=j ; lanes16-31: M=j+8.
// ---------------------------------------------------------------------------
__global__ void gcn_mlp_wmma_kernel(const float* __restrict__ x,
                                    const float* __restrict__ W1,
                                    const float* __restrict__ b1,
                                    const float* __restrict__ W2,
                                    const float* __restrict__ b2,
                                    float* __restrict__ h_out,
                                    float* __restrict__ out_pairs,
                                    int n_nodes, int n_tiles) {
    int gtid = blockIdx.x * blockDim.x + threadIdx.x;

    // zero (agg, cnt) pairs in d_out (harness poisons it with 0xAA)
    if (gtid < n_nodes) {
        ((float2*)out_pairs)[gtid] = make_float2(0.0f, 0.0f);
    }

    int wave = gtid >> 5;
    int lane = gtid & 31;
    if (wave >= n_tiles) return;       // wave-uniform: EXEC all-ones below

    int m  = lane & 15;                // row within 16-node tile
    int hi = lane >> 4;                // 0 -> K slots {kb,kb+1}; 1 -> +2
    int node = wave * 16 + m;
    float xv = x[node < n_nodes ? node : (n_nodes - 1)];

    v8f acc = {};
#pragma unroll
    for (int c = 0; c < 4; ++c) {
        int kb = 4 * c + 2 * hi;
        v2f a, b;
        a.x = fmaxf(fmaf(xv, W1[kb],     b1[kb]),     0.0f);
        a.y = fmaxf(fmaf(xv, W1[kb + 1], b1[kb + 1]), 0.0f);
        b.x = W2[kb];
        b.y = W2[kb + 1];
        // 8 args: (neg_a, A, neg_b, B, c_mod, C, reuse_a, reuse_b)
        acc = __builtin_amdgcn_wmma_f32_16x16x4_f32(
                  false, a, false, b, (short)0, acc, false, false);
    }

    // One lane per half-wave writes its 8 rows of h. base is 32B-aligned, so
    // the full-tile path is two global_store_b128 (all tiles are full when
    // n_nodes % 16 == 0, as here: 100000 = 6250 * 16).
    if (m == 0) {
        float bias2 = b2[0];
        int base = wave * 16 + hi * 8;
        if (base + 8 <= n_nodes) {
            float4 lo = make_float4(acc[0] + bias2, acc[1] + bias2,
                                    acc[2] + bias2, acc[3] + bias2);
            float4 hi4 = make_float4(acc[4] + bias2, acc[5] + bias2,
                                     acc[6] + bias2, acc[7] + bias2);
            *(float4*)(h_out + base)     = lo;
            *(float4*)(h_out + base + 4) = hi4;
        } else {                        // generic partial-tile tail (unused here)
            float t[8] = {acc[0], acc[1], acc[2], acc[3],
                          acc[4], acc[5], acc[6], acc[7]};
#pragma unroll
            for (int j = 0; j < 8; ++j)
                if (base + j < n_nodes) h_out[base + j] = t[j] + bias2;
        }
    }
}

// ---------------------------------------------------------------------------
// Kernel B: stream 6.4M edges (102.4 MB of int64 indices -> HBM-bound floor
// ~4.5us at 23.3 TB/s), gather h[src] (L2-resident, 400 KB), scatter-add
// (h, 1.0) into the interleaved (agg, cnt) pair at out[2*dst] with native
// non-returning f32 L2 atomics. 4 edges per thread per iteration via four
// 16-byte coalesced loads (512B/wave per load instruction).
// ---------------------------------------------------------------------------
__global__ void gcn_edge_kernel(const long long* __restrict__ src,
                                const long long* __restrict__ dst,
                                const float* __restrict__ h,
                                float* __restrict__ out_pairs,
                                long long n_edges) {
    long long quads  = n_edges >> 2;          // groups of 4 edges
    long long tid    = (long long)blockIdx.x * blockDim.x + threadIdx.x;
    long long stride = (long long)gridDim.x * blockDim.x;
    const LL2* s2 = (const LL2*)src;
    const LL2* d2 = (const LL2*)dst;

    for (long long i = tid; i < quads; i += stride) {
        LL2 sa = s2[2 * i];
        LL2 sb = s2[2 * i + 1];
        LL2 da = d2[2 * i];
        LL2 db = d2[2 * i + 1];
        int s0 = (int)sa.x, s1 = (int)sa.y;   // indices < 100000: low 32 bits
        int s2i = (int)sb.x, s3 = (int)sb.y;
        int d0 = (int)da.x, d1 = (int)da.y;
        int d2i = (int)db.x, d3 = (int)db.y;
        float h0 = h[s0];
        float h1 = h[s1];
        float h2 = h[s2i];
        float h3 = h[s3];
        unsafeAtomicAdd(&out_pairs[2 * d0],      h0);
        unsafeAtomicAdd(&out_pairs[2 * d0 + 1],  1.0f);
        unsafeAtomicAdd(&out_pairs[2 * d1],      h1);
        unsafeAtomicAdd(&out_pairs[2 * d1 + 1],  1.0f);
        unsafeAtomicAdd(&out_pairs[2 * d2i],     h2);
        unsafeAtomicAdd(&out_pairs[2 * d2i + 1], 1.0f);
        unsafeAtomicAdd(&out_pairs[2 * d3],      h3);
        unsafeAtomicAdd(&out_pairs[2 * d3 + 1],  1.0f);
    }

    // generic tail for n_edges % 4 != 0 (unused for E = 6.4M)
    long long tail_base = quads << 2;
    long long t = tail_base + tid;
    if (t < n_edges) {
        int s0 = (int)src[t];
        int d0 = (int)dst[t];
        unsafeAtomicAdd(&out_pairs[2 * d0],     h[s0]);
        unsafeAtomicAdd(&out_pairs[2 * d0 + 1], 1.0f);
    }
}

// ---------------------------------------------------------------------------
// Kernel C: in-place finalize: mean = agg / max(cnt,1);  out[i,:] = mean * Ws
// ---------------------------------------------------------------------------
__global__ void gcn_finalize_kernel(float* __restrict__ out,
                                    const float* __restrict__ Ws,
                                    int n_nodes) {
    int i = blockIdx.x * blockDim.x + threadIdx.x;
    if (i >= n_nodes) return;
    float2 p = ((float2*)out)[i];
    float mean = p.x / fmaxf(p.y, 1.0f);
    ((float2*)out)[i] = make_float2(mean * Ws[0], mean * Ws[1]);
}

extern "C" void kernel_launch(void* const* d_in, const int* in_sizes, int n_in,
                              void* d_out, int out_size, void* d_ws, size_t ws_size,
                              hipStream_t stream) {
    const float*     x  = (const float*)d_in[0];
    const float*     W1 = (const float*)d_in[1];
    const float*     b1 = (const float*)d_in[2];
    const float*     W2 = (const float*)d_in[3];
    const float*     b2 = (const float*)d_in[4];
    const float*     Ws = (const float*)d_in[5];
    const long long* ei = (const long long*)d_in[6];

    int       n_nodes = in_sizes[0];                  // 100000
    long long n_edges = (long long)in_sizes[6] / 2;   // 6400000
    const long long* src = ei;
    const long long* dst = ei + n_edges;

    float* out = (float*)d_out;   // interleaved (agg, cnt) then final [N,2]
    float* h   = (float*)d_ws;    // 400 KB scratch

    // A: one wave per 16-node tile (100000/16 = 6250 full tiles) + zeroing
    int tiles    = (n_nodes + 15) / 16;
    int threadsA = tiles * 32;
    int blocksA  = (threadsA + 255) / 256;
    gcn_mlp_wmma_kernel<<<blocksA, 256, 0, stream>>>(x, W1, b1, W2, b2,
                                                     h, out, n_nodes, tiles);

    // B: 524288 threads grid-striding over 1.6M four-edge packets
    gcn_edge_kernel<<<2048, 256, 0, stream>>>(src, dst, h, out, n_edges);

    // C: finalize in place
    int blocksC = (n_nodes + 255) / 256;
    gcn_finalize_kernel<<<blocksC, 256, 0, stream>>>(out, Ws, n_nodes);
}